// EdgeHead_64080912056779
// MI455X (gfx1250) — compile-verified
//
#include <hip/hip_runtime.h>
#include <hip/hip_bf16.h>

typedef __attribute__((ext_vector_type(2))) float v2f;
typedef __attribute__((ext_vector_type(4))) float v4f;
typedef __attribute__((ext_vector_type(8))) float v8f;

#define C_DIM      128
#define TILE_M     128
#define SA_STRIDE  132               // 128 + 4 pad: A-frag column reads hit distinct banks
#define SBP_STRIDE 288               // dwords per K-pair row; 288 % 64 == 32 -> conflict-free b64 frags
#define SA_FLOATS  (TILE_M * SA_STRIDE)            // 16896
#define SB_FLOATS  ((C_DIM / 2) * SBP_STRIDE)      // 64 K-pairs * 288 = 18432
#define SMEM_BYTES ((SA_FLOATS + SB_FLOATS) * 4)   // 141312 B < 320KB/WGP

__global__ __launch_bounds__(256, 1)
void edge_mlp_wmma_f32(const float* __restrict__ x,
                       const long long* __restrict__ eidx,
                       const float* __restrict__ W1,
                       const float* __restrict__ b1,
                       const float* __restrict__ W2,
                       const float* __restrict__ b2,
                       float* __restrict__ out,
                       int nEdges)
{
    extern __shared__ float smem[];
    float* sA = smem;              // gathered edge tile [128][132]
    float* sB = smem + SA_FLOATS;  // W1 interleaved: (k,n) -> [(k>>1)*288 + 2n + (k&1)]

    const int tid = threadIdx.x;
    const long long e0 = (long long)blockIdx.x * TILE_M;

    const long long* heads = eidx;
    const long long* tails = eidx + nEdges;

    // ---- Stage A: gather x[head]+x[tail] for 128 edges into LDS ----
    {
        const int r    = tid >> 1;         // edge row within tile, 0..127
        const int half = (tid & 1) * 64;   // column half, 0 or 64
        float* rowA = &sA[r * SA_STRIDE + half];
        const long long e = e0 + r;
        if (e < nEdges) {
            const long long h  = heads[e];
            const long long t2 = tails[e];
            const v4f* xh = (const v4f*)(x + (size_t)h  * C_DIM + half);
            const v4f* xt = (const v4f*)(x + (size_t)t2 * C_DIM + half);
            #pragma unroll
            for (int i = 0; i < 16; ++i) {
                v4f v = xh[i] + xt[i];
                *(v4f*)&rowA[i * 4] = v;
            }
        } else {
            const v4f z = {};
            #pragma unroll
            for (int i = 0; i < 16; ++i) *(v4f*)&rowA[i * 4] = z;
        }
    }

    // ---- Stage B: W1 into LDS, interleaved by K-pair so each B fragment
    //      is one aligned 8-byte load into an even VGPR pair ----
    {
        const int r    = tid >> 1;         // W1 row (K index), 0..127
        const int half = (tid & 1) * 64;
        const int parity = r & 1;
        float* pairRow = &sB[(r >> 1) * SBP_STRIDE + parity];
        const v4f* w1v = (const v4f*)(W1 + (size_t)r * C_DIM + half);
        #pragma unroll
        for (int i = 0; i < 16; ++i) {
            const v4f v = w1v[i];
            const int c = half + i * 4;
            pairRow[2 * (c + 0)] = v.x;
            pairRow[2 * (c + 1)] = v.y;
            pairRow[2 * (c + 2)] = v.z;
            pairRow[2 * (c + 3)] = v.w;
        }
    }

    __syncthreads();

    // ---- GEMM1 via fp32 WMMA: each wave owns 16 rows x 128 cols ----
    const int lane = tid & 31;
    const int wid  = tid >> 5;        // 0..7
    const int m0   = wid * 16;
    const int ln   = lane & 15;       // N within tile / M within A frag
    const int kh   = (lane >> 4) * 2; // K sub-offset per lane half (0 or 2)

    const v8f vzero = {};
    v8f acc[8];
    #pragma unroll
    for (int t = 0; t < 8; ++t) {
        // C/D layout: value at [M][N=t*16+ln]; bias depends only on N
        acc[t] = vzero + b1[t * 16 + ln];
    }

    const float* aRow  = &sA[(m0 + ln) * SA_STRIDE + kh];
    const float* bBase = &sB[(kh >> 1) * SBP_STRIDE + 2 * ln];
    for (int k = 0; k < C_DIM; k += 4) {
        // A frag (16x4 f32): lane holds row m0+ln, K = k+kh, k+kh+1 (contiguous)
        const v2f a = *(const v2f*)&aRow[k];
        const float* bK = &bBase[(k >> 1) * SBP_STRIDE];
        #pragma unroll
        for (int t = 0; t < 8; ++t) {
            // B frag: (K=k+kh, n) and (K=k+kh+1, n) adjacent in LDS
            const v2f b = *(const v2f*)&bK[t * 32];
            acc[t] = __builtin_amdgcn_wmma_f32_16x16x4_f32(
                false, a, false, b, (short)0, acc[t], false, false);
        }
    }

    // ---- Epilogue: relu, GEMM2 (N=2) as dot + 16-lane butterfly ----
    float w2a[8], w2b[8];
    #pragma unroll
    for (int t = 0; t < 8; ++t) {
        const int n = t * 16 + ln;
        w2a[t] = W2[n * 2 + 0];
        w2b[t] = W2[n * 2 + 1];
    }
    const float b2v0 = b2[0];
    const float b2v1 = b2[1];

    #pragma unroll
    for (int r = 0; r < 8; ++r) {
        float p0 = 0.f, p1 = 0.f;
        #pragma unroll
        for (int t = 0; t < 8; ++t) {
            float hv = acc[t][r];
            hv = hv > 0.f ? hv : 0.f;      // relu
            p0 += hv * w2a[t];
            p1 += hv * w2b[t];
        }
        // reduce over N (lanes within each 16-lane half)
        #pragma unroll
        for (int off = 8; off; off >>= 1) {
            p0 += __shfl_xor(p0, off, 32);
            p1 += __shfl_xor(p1, off, 32);
        }
        if (ln == 0) {
            const int m = m0 + r + ((lane >> 4) << 3); // lane half selects M / M+8
            const long long e = e0 + m;
            if (e < nEdges) {
                out[2 * e + 0] = p0 + b2v0;
                out[2 * e + 1] = p1 + b2v1;
            }
        }
    }
}

extern "C" void kernel_launch(void* const* d_in, const int* in_sizes, int n_in,
                              void* d_out, int out_size, void* d_ws, size_t ws_size,
                              hipStream_t stream) {
    const float*     x    = (const float*)d_in[0];
    const long long* eidx = (const long long*)d_in[1];
    const float*     W1   = (const float*)d_in[2];
    const float*     b1   = (const float*)d_in[3];
    const float*     W2   = (const float*)d_in[4];
    const float*     b2   = (const float*)d_in[5];
    float*           out  = (float*)d_out;

    const int nEdges = in_sizes[1] / 2;
    const int nBlocks = (nEdges + TILE_M - 1) / TILE_M;

    static_assert(SMEM_BYTES <= 320 * 1024, "LDS per WGP exceeded");
    hipFuncSetAttribute((const void*)edge_mlp_wmma_f32,
                        hipFuncAttributeMaxDynamicSharedMemorySize, SMEM_BYTES);

    edge_mlp_wmma_f32<<<nBlocks, 256, SMEM_BYTES, stream>>>(
        x, eidx, W1, b1, W2, b2, out, nEdges);
}